// EdgeUpdate_9990093930531
// MI455X (gfx1250) — compile-verified
//
#include <hip/hip_runtime.h>
#include <math.h>

typedef float v2f __attribute__((ext_vector_type(2)));
typedef float v8f __attribute__((ext_vector_type(8)));

#define HN 64
#define HE 64
#define NN 50000
#define EE 200000
#define TT 600000
#define EPSV 1e-5f

// ---------------- helpers ----------------
__device__ __forceinline__ void stage16(float* dst, const float* src) {
  const float4* s = (const float4*)src;
  float4* d = (float4*)dst;
#pragma unroll
  for (int q = 0; q < 4; ++q) d[q] = s[q];
}
__device__ __forceinline__ void stage16_mul(float* dst, const float* sa, const float* sb) {
  const float4* a4 = (const float4*)sa;
  const float4* b4 = (const float4*)sb;
  float4* d = (float4*)dst;
#pragma unroll
  for (int q = 0; q < 4; ++q) {
    float4 x = a4[q], y = b4[q];
    d[q] = make_float4(x.x*y.x, x.y*y.y, x.z*y.z, x.w*y.w);
  }
}
__device__ __forceinline__ float sigmoidf_(float x) { return 1.0f / (1.0f + __expf(-x)); }

// epilogue for one 16x16 f32 C/D tile: add bias, store, reduce col stats.
// C/D layout: lane n (n<16) holds rows 0..7 of col c in acc[0..7]; lane n+16 holds rows 8..15.
__device__ __forceinline__ void epilogue_tile(v8f acc, int c, int row_base, float bias,
                                              float* __restrict__ Yout,
                                              float* s_sum, float* s_sq, int hi) {
  float ps = 0.f, pq = 0.f;
#pragma unroll
  for (int v = 0; v < 8; ++v) {
    float y = acc[v] + bias;
    Yout[(size_t)(row_base + v) * 128 + c] = y;
    ps += y; pq += y * y;
  }
  ps += __shfl_xor(ps, 16, 32);   // combine rows 0-7 with rows 8-15 halves
  pq += __shfl_xor(pq, 16, 32);
  if (hi == 0) { atomicAdd(&s_sum[c], ps); atomicAdd(&s_sq[c], pq); }
}

// ---------------- utility kernels ----------------
__global__ void k_zero(float* __restrict__ p, long long n) {
  long long i = (long long)blockIdx.x * blockDim.x + threadIdx.x;
  long long st = (long long)gridDim.x * blockDim.x;
  for (; i < n; i += st) p[i] = 0.f;
}

// W (C x K, row-major) -> paired layout Wp[(k>>1)*C + c][2] = {W[c][k_even], W[c][k_odd]}
__global__ void k_tp(float* __restrict__ dst, const float* __restrict__ src, int K, int C) {
  int t = blockIdx.x * blockDim.x + threadIdx.x;
  if (t >= K * C) return;
  int k = t / C, c = t - k * C;
  dst[((size_t)(k >> 1) * C + c) * 2 + (k & 1)] = src[(size_t)c * K + k];
}

// ---------------- GEMM 2: Y2[E x 128] = (node[i]*node[j]) @ W2^T + b2, + col stats ----------------
__global__ __launch_bounds__(256) void k_gemm2(
    const float* __restrict__ node_emb, const int* __restrict__ ii, const int* __restrict__ jj,
    const float* __restrict__ Wt2p, const float* __restrict__ b2,
    float* __restrict__ Y2, float* __restrict__ sums, float* __restrict__ sumsq) {
  __shared__ float A[64][68];          // pad 68: bank = (4r + k) % 64, conflict-free
  __shared__ float s_sum[128], s_sq[128];
  const int tid = threadIdx.x;
  const int r0 = blockIdx.x * 64;
  {
    const int r = tid >> 2, part = tid & 3;
    const int e = r0 + r;
    stage16_mul(&A[r][part * 16],
                node_emb + (size_t)ii[e] * HN + part * 16,
                node_emb + (size_t)jj[e] * HN + part * 16);
  }
  if (tid < 128) { s_sum[tid] = 0.f; s_sq[tid] = 0.f; }
  __syncthreads();

  const int wave = tid >> 5, lane = tid & 31;
  const int rt = wave >> 1, chalf = wave & 1;
  const int ln = lane & 15, hi = lane >> 4;
  const int arow = rt * 16 + ln;

  v8f acc0 = {}, acc1 = {}, acc2 = {}, acc3 = {};
#pragma unroll
  for (int k0 = 0; k0 < 64; k0 += 4) {
    // A frag (16x4 f32): lane<16 -> (K=k0,k0+1), lane>=16 -> (K=k0+2,k0+3)
    v2f a = *(const v2f*)&A[arow][k0 + 2 * hi];
    const float* bp = Wt2p + ((size_t)((k0 >> 1) + hi) * 128 + chalf * 64 + ln) * 2;
    v2f b0v = *(const v2f*)(bp);
    v2f b1v = *(const v2f*)(bp + 32);
    v2f b2v = *(const v2f*)(bp + 64);
    v2f b3v = *(const v2f*)(bp + 96);
    acc0 = __builtin_amdgcn_wmma_f32_16x16x4_f32(false, a, false, b0v, (short)0, acc0, false, false);
    acc1 = __builtin_amdgcn_wmma_f32_16x16x4_f32(false, a, false, b1v, (short)0, acc1, false, false);
    acc2 = __builtin_amdgcn_wmma_f32_16x16x4_f32(false, a, false, b2v, (short)0, acc2, false, false);
    acc3 = __builtin_amdgcn_wmma_f32_16x16x4_f32(false, a, false, b3v, (short)0, acc3, false, false);
  }
  const int row_base = r0 + rt * 16 + hi * 8;
  {
    int c = chalf * 64 + ln;
    epilogue_tile(acc0, c +  0, row_base, b2[c +  0], Y2, s_sum, s_sq, hi);
    epilogue_tile(acc1, c + 16, row_base, b2[c + 16], Y2, s_sum, s_sq, hi);
    epilogue_tile(acc2, c + 32, row_base, b2[c + 32], Y2, s_sum, s_sq, hi);
    epilogue_tile(acc3, c + 48, row_base, b2[c + 48], Y2, s_sum, s_sq, hi);
  }
  __syncthreads();
  if (tid < 128) {
    unsafeAtomicAdd(&sums[tid], s_sum[tid]);
    unsafeAtomicAdd(&sumsq[tid], s_sq[tid]);
  }
}

// ---------------- GEMM 3: Y3[T x 128] = concat5 @ W3^T + b3, gather fused, + col stats ----------------
__global__ __launch_bounds__(256) void k_gemm3(
    const float* __restrict__ node_emb, const float* __restrict__ edge_emb,
    const int* __restrict__ idx_i, const int* __restrict__ idx_j, const int* __restrict__ idx_k,
    const int* __restrict__ idx_ji, const int* __restrict__ idx_kj,
    const float* __restrict__ Wt3p, const float* __restrict__ b3,
    float* __restrict__ Y3, float* __restrict__ sums, float* __restrict__ sumsq) {
  __shared__ float A[64][324];         // ~83 KB of the 320 KB WGP LDS; pad 324 -> conflict-free
  __shared__ float s_sum[128], s_sq[128];
  const int tid = threadIdx.x;
  const int r0 = blockIdx.x * 64;
  {
    const int r = tid >> 2, part = tid & 3;
    const int t = r0 + r;
    stage16(&A[r][0 * 64 + part * 16], node_emb + (size_t)idx_i[t]  * HN + part * 16);
    stage16(&A[r][1 * 64 + part * 16], node_emb + (size_t)idx_j[t]  * HN + part * 16);
    stage16(&A[r][2 * 64 + part * 16], node_emb + (size_t)idx_k[t]  * HN + part * 16);
    stage16(&A[r][3 * 64 + part * 16], edge_emb + (size_t)idx_ji[t] * HE + part * 16);
    stage16(&A[r][4 * 64 + part * 16], edge_emb + (size_t)idx_kj[t] * HE + part * 16);
  }
  if (tid < 128) { s_sum[tid] = 0.f; s_sq[tid] = 0.f; }
  __syncthreads();

  const int wave = tid >> 5, lane = tid & 31;
  const int rt = wave >> 1, chalf = wave & 1;
  const int ln = lane & 15, hi = lane >> 4;
  const int arow = rt * 16 + ln;

  v8f acc0 = {}, acc1 = {}, acc2 = {}, acc3 = {};
#pragma unroll 4
  for (int k0 = 0; k0 < 320; k0 += 4) {
    v2f a = *(const v2f*)&A[arow][k0 + 2 * hi];
    const float* bp = Wt3p + ((size_t)((k0 >> 1) + hi) * 128 + chalf * 64 + ln) * 2;
    v2f b0v = *(const v2f*)(bp);
    v2f b1v = *(const v2f*)(bp + 32);
    v2f b2v = *(const v2f*)(bp + 64);
    v2f b3v = *(const v2f*)(bp + 96);
    acc0 = __builtin_amdgcn_wmma_f32_16x16x4_f32(false, a, false, b0v, (short)0, acc0, false, false);
    acc1 = __builtin_amdgcn_wmma_f32_16x16x4_f32(false, a, false, b1v, (short)0, acc1, false, false);
    acc2 = __builtin_amdgcn_wmma_f32_16x16x4_f32(false, a, false, b2v, (short)0, acc2, false, false);
    acc3 = __builtin_amdgcn_wmma_f32_16x16x4_f32(false, a, false, b3v, (short)0, acc3, false, false);
  }
  const int row_base = r0 + rt * 16 + hi * 8;
  {
    int c = chalf * 64 + ln;
    epilogue_tile(acc0, c +  0, row_base, b3[c +  0], Y3, s_sum, s_sq, hi);
    epilogue_tile(acc1, c + 16, row_base, b3[c + 16], Y3, s_sum, s_sq, hi);
    epilogue_tile(acc2, c + 32, row_base, b3[c + 32], Y3, s_sum, s_sq, hi);
    epilogue_tile(acc3, c + 48, row_base, b3[c + 48], Y3, s_sum, s_sq, hi);
  }
  __syncthreads();
  if (tid < 128) {
    unsafeAtomicAdd(&sums[tid], s_sum[tid]);
    unsafeAtomicAdd(&sumsq[tid], s_sq[tid]);
  }
}

// ---------------- BN(c2) + sigmoid*tanh -> act2, + act2 col stats ----------------
__global__ __launch_bounds__(256) void k_bnact2(
    const float* __restrict__ Y2, const float* __restrict__ sums2, const float* __restrict__ sumsq2,
    const float* __restrict__ g2, const float* __restrict__ be2,
    float* __restrict__ act2, float* __restrict__ s22, float* __restrict__ q22) {
  __shared__ float ss[64], sq[64];
  const int tid = threadIdx.x;
  const int c = tid & 63, rs = tid >> 6;
  const int e = blockIdx.x * 4 + rs;
  if (tid < 64) { ss[tid] = 0.f; sq[tid] = 0.f; }
  __syncthreads();
  const float invE = 1.0f / (float)EE;
  const float mf = sums2[c] * invE,       vf = sumsq2[c] * invE - mf * mf;
  const float mc = sums2[c + 64] * invE,  vc = sumsq2[c + 64] * invE - mc * mc;
  const float zf = g2[c]      * (Y2[(size_t)e * 128 + c]      - mf) * rsqrtf(vf + EPSV) + be2[c];
  const float zc = g2[c + 64] * (Y2[(size_t)e * 128 + c + 64] - mc) * rsqrtf(vc + EPSV) + be2[c + 64];
  const float m = sigmoidf_(zf) * tanhf(zc);
  act2[(size_t)e * 64 + c] = m;
  atomicAdd(&ss[c], m); atomicAdd(&sq[c], m * m);
  __syncthreads();
  if (tid < 64) { unsafeAtomicAdd(&s22[tid], ss[tid]); unsafeAtomicAdd(&q22[tid], sq[tid]); }
}

// ---------------- BN(c3) + sigmoid*tanh + segment-sum scatter ----------------
__global__ __launch_bounds__(256) void k_msg(
    const float* __restrict__ Y3, const float* __restrict__ sums3, const float* __restrict__ sumsq3,
    const float* __restrict__ g3, const float* __restrict__ be3,
    const int* __restrict__ idx_ji, float* __restrict__ agg) {
  const int tid = threadIdx.x;
  const int c = tid & 63, rs = tid >> 6;
  const int t = blockIdx.x * 4 + rs;
  const float invT = 1.0f / (float)TT;
  const float mf = sums3[c] * invT,       vf = sumsq3[c] * invT - mf * mf;
  const float mc = sums3[c + 64] * invT,  vc = sumsq3[c + 64] * invT - mc * mc;
  const float zf = g3[c]      * (Y3[(size_t)t * 128 + c]      - mf) * rsqrtf(vf + EPSV) + be3[c];
  const float zc = g3[c + 64] * (Y3[(size_t)t * 128 + c + 64] - mc) * rsqrtf(vc + EPSV) + be3[c + 64];
  const float m = sigmoidf_(zf) * tanhf(zc);
  unsafeAtomicAdd(agg + (size_t)idx_ji[t] * 64 + c, m);   // global_atomic_add_f32
}

// ---------------- column stats over a (nrows x 64) matrix ----------------
__global__ __launch_bounds__(256) void k_colstats64(
    const float* __restrict__ X, int nrows, float* __restrict__ sums, float* __restrict__ sumsq) {
  __shared__ float ss[64], sq[64];
  const int tid = threadIdx.x;
  const int c = tid & 63, rs = tid >> 6;
  if (tid < 64) { ss[tid] = 0.f; sq[tid] = 0.f; }
  __syncthreads();
  float s = 0.f, q = 0.f;
  for (int r = blockIdx.x * 4 + rs; r < nrows; r += gridDim.x * 4) {
    float x = X[(size_t)r * 64 + c];
    s += x; q += x * x;
  }
  atomicAdd(&ss[c], s); atomicAdd(&sq[c], q);
  __syncthreads();
  if (tid < 64) { unsafeAtomicAdd(&sums[tid], ss[tid]); unsafeAtomicAdd(&sumsq[tid], sq[tid]); }
}

// ---------------- final: tanh(edge + BN(act2) + BN(agg)) ----------------
__global__ __launch_bounds__(256) void k_final(
    const float* __restrict__ edge_emb,
    const float* __restrict__ act2, const float* __restrict__ s22, const float* __restrict__ q22,
    const float* __restrict__ g22, const float* __restrict__ be22,
    const float* __restrict__ agg, const float* __restrict__ s32, const float* __restrict__ q32,
    const float* __restrict__ g32, const float* __restrict__ be32,
    float* __restrict__ out) {
  const int tid = threadIdx.x;
  const int c = tid & 63, rs = tid >> 6;
  const int e = blockIdx.x * 4 + rs;
  const float invE = 1.0f / (float)EE;
  const float m2 = s22[c] * invE, v2 = q22[c] * invE - m2 * m2;
  const float m3 = s32[c] * invE, v3 = q32[c] * invE - m3 * m3;
  const float a2 = g22[c] * (act2[(size_t)e * 64 + c] - m2) * rsqrtf(v2 + EPSV) + be22[c];
  const float a3 = g32[c] * (agg [(size_t)e * 64 + c] - m3) * rsqrtf(v3 + EPSV) + be32[c];
  out[(size_t)e * 64 + c] = tanhf(edge_emb[(size_t)e * 64 + c] + a2 + a3);
}

// ---------------- launch ----------------
extern "C" void kernel_launch(void* const* d_in, const int* in_sizes, int n_in,
                              void* d_out, int out_size, void* d_ws, size_t ws_size,
                              hipStream_t stream) {
  (void)in_sizes; (void)n_in; (void)out_size; (void)ws_size;
  const float* node_emb = (const float*)d_in[0];
  const float* edge_emb = (const float*)d_in[1];
  const int*   i_idx    = (const int*)d_in[2];
  const int*   j_idx    = (const int*)d_in[3];
  const int*   idx_i    = (const int*)d_in[4];
  const int*   idx_j    = (const int*)d_in[5];
  const int*   idx_k    = (const int*)d_in[6];
  const int*   idx_ji   = (const int*)d_in[7];
  const int*   idx_kj   = (const int*)d_in[8];
  const float* W2       = (const float*)d_in[9];
  const float* b2       = (const float*)d_in[10];
  const float* W3       = (const float*)d_in[11];
  const float* b3       = (const float*)d_in[12];
  const float* g_c2     = (const float*)d_in[13];
  const float* beta_c2  = (const float*)d_in[14];
  const float* g_c3     = (const float*)d_in[15];
  const float* beta_c3  = (const float*)d_in[16];
  const float* g_c22    = (const float*)d_in[17];
  const float* beta_c22 = (const float*)d_in[18];
  const float* g_c32    = (const float*)d_in[19];
  const float* beta_c32 = (const float*)d_in[20];

  float* ws   = (float*)d_ws;
  float* Y3   = ws;                             // T*128 (Y2 aliases same region; lifetimes disjoint)
  float* Y2   = ws;                             // E*128
  float* act2 = ws + (size_t)TT * 128;          // E*64
  float* agg  = act2 + (size_t)EE * 64;         // E*64
  float* Wt2p = agg + (size_t)EE * 64;          // 64*128
  float* Wt3p = Wt2p + 64 * 128;                // 320*128
  float* stats = Wt3p + 320 * 128;              // 768 floats
  // stats layout: [0]=sum2(128) [128]=sq2 [256]=sum3(128) [384]=sq3
  //               [512]=sum22(64) [576]=sq22 [640]=sum32(64) [704]=sq32

  k_zero<<<4096, 256, 0, stream>>>(agg, (long long)EE * 64);
  k_zero<<<4, 256, 0, stream>>>(stats, 768);
  k_tp<<<(64 * 128 + 255) / 256, 256, 0, stream>>>(Wt2p, W2, 64, 128);
  k_tp<<<(320 * 128 + 255) / 256, 256, 0, stream>>>(Wt3p, W3, 320, 128);

  k_gemm2<<<EE / 64, 256, 0, stream>>>(node_emb, i_idx, j_idx, Wt2p, b2,
                                       Y2, stats + 0, stats + 128);
  k_bnact2<<<EE / 4, 256, 0, stream>>>(Y2, stats + 0, stats + 128, g_c2, beta_c2,
                                       act2, stats + 512, stats + 576);
  k_gemm3<<<TT / 64, 256, 0, stream>>>(node_emb, edge_emb, idx_i, idx_j, idx_k, idx_ji, idx_kj,
                                       Wt3p, b3, Y3, stats + 256, stats + 384);
  k_msg<<<TT / 4, 256, 0, stream>>>(Y3, stats + 256, stats + 384, g_c3, beta_c3, idx_ji, agg);
  k_colstats64<<<2048, 256, 0, stream>>>(agg, EE, stats + 640, stats + 704);
  k_final<<<EE / 4, 256, 0, stream>>>(edge_emb, act2, stats + 512, stats + 576, g_c22, beta_c22,
                                      agg, stats + 640, stats + 704, g_c32, beta_c32,
                                      (float*)d_out);
}